// SpatialVisionAggregator_77902116814880
// MI455X (gfx1250) — compile-verified
//
#include <hip/hip_runtime.h>
#include <hip/hip_bf16.h>

// CDNA5 / gfx1250, wave32. Uses native f32 WMMA: V_WMMA_F32_16X16X4_F32.
// D = A(16x4) x B(4x16) + C(16x16), all f32.
//
// VGPR layouts (ISA 7.12.2, 32-bit):
//   A: lane L, VGPR j  -> A[L%16][2*(L/16)+j]
//   B: lane L, VGPR j  -> B[2*(L/16)+j][L%16]
//   C/D: lane L, VGPR r -> D[r + 8*(L/16)][L%16]

typedef float v2f __attribute__((ext_vector_type(2)));
typedef float v8f __attribute__((ext_vector_type(8)));

#define WMMA_F32(acc, a, b) \
  (acc) = __builtin_amdgcn_wmma_f32_16x16x4_f32(false, (a), false, (b), (short)0, (acc), false, false)

// Problem constants
#define BB 8
#define L_KV 1600
#define DD 2048
#define TT 729          // 27*27 queries
#define SIDE 41         // kv grid side
#define NEGBIG -1.0e9f

// ---------------------------------------------------------------------------
// Kernel 1: kv[b,l,d] = (l<576 ? x0[b,l,:]@W0[d,:] + b0[d]
//                               : x1[b,l-576,:]@W1[d,:] + b1[d])
// One wave per block; each wave computes a 16(l) x 64(d) tile (4 WMMA accums).
// grid = (DD/64=32, L_KV/16=100, BB=8), block = 32
// ---------------------------------------------------------------------------
__global__ __launch_bounds__(32)
void kv_proj_kernel(const float* __restrict__ x0, const float* __restrict__ x1,
                    const float* __restrict__ W0, const float* __restrict__ b0v,
                    const float* __restrict__ W1, const float* __restrict__ b1v,
                    float* __restrict__ kv)
{
    const int lane = threadIdx.x;
    const int m = lane & 15;
    const int h = lane >> 4;
    const int n0 = blockIdx.x * 64;
    const int lt = blockIdx.y;
    const int b  = blockIdx.z;

    const float* xrow;
    const float* Wm;
    const float* bias;
    int K, lbase;
    if (lt < 36) {                       // x0 section: rows 0..575, K=1024
        K = 1024; Wm = W0; bias = b0v;
        lbase = lt * 16;
        xrow = x0 + ((size_t)b * 576 + lbase + m) * 1024;
    } else {                             // x1 section: rows 576..1599, K=1536
        K = 1536; Wm = W1; bias = b1v;
        lbase = 576 + (lt - 36) * 16;
        xrow = x1 + ((size_t)b * 1024 + (lt - 36) * 16 + m) * 1536;
    }

    const float* w0p = Wm + (size_t)(n0 +  0 + m) * K;
    const float* w1p = Wm + (size_t)(n0 + 16 + m) * K;
    const float* w2p = Wm + (size_t)(n0 + 32 + m) * K;
    const float* w3p = Wm + (size_t)(n0 + 48 + m) * K;

    v8f acc0 = {}, acc1 = {}, acc2 = {}, acc3 = {};
    for (int kb = 0; kb < K; kb += 4) {
        const int ko = kb + 2 * h;
        v2f a  = *(const v2f*)(xrow + ko);
        v2f q0 = *(const v2f*)(w0p + ko);
        v2f q1 = *(const v2f*)(w1p + ko);
        v2f q2 = *(const v2f*)(w2p + ko);
        v2f q3 = *(const v2f*)(w3p + ko);
        WMMA_F32(acc0, a, q0);
        WMMA_F32(acc1, a, q1);
        WMMA_F32(acc2, a, q2);
        WMMA_F32(acc3, a, q3);
    }

    const float bi0 = bias[n0 +  0 + m];
    const float bi1 = bias[n0 + 16 + m];
    const float bi2 = bias[n0 + 32 + m];
    const float bi3 = bias[n0 + 48 + m];
#pragma unroll
    for (int r = 0; r < 8; ++r) {
        const int l = lbase + r + 8 * h;
        float* orow = kv + ((size_t)b * L_KV + l) * DD;
        orow[n0 +  0 + m] = acc0[r] + bi0;
        orow[n0 + 16 + m] = acc1[r] + bi1;
        orow[n0 + 32 + m] = acc2[r] + bi2;
        orow[n0 + 48 + m] = acc3[r] + bi3;
    }
}

// ---------------------------------------------------------------------------
// Kernel 2 (per batch): scores[t,l] = scale * q[t,:]·kv[b,l,:] + bias(t,l)
// Each wave: 16(t) x 64(l) tile. grid = (L_KV/64=25, ceil(729/16)=46), block=32
// ---------------------------------------------------------------------------
__device__ __forceinline__ float spatial_bias(int t, int l, float qy, float qx,
                                              float lo, float hi)
{
    const float ly = (float)(l / SIDE) * (1.0f / 40.0f);
    const float lx = (float)(l % SIDE) * (1.0f / 40.0f);  // == kv_y in reference
    const float dy = qy - ly;
    const float dx = qx - lx;
    const float dist = sqrtf(dy * dy + dx * dx);
    const bool mask = (lx >= lo) && (lx < hi);
    return mask ? -dist : NEGBIG;
}

__global__ __launch_bounds__(32)
void scores_kernel(const float* __restrict__ lq, const float* __restrict__ kv,
                   float* __restrict__ scores, int b)
{
    const int lane = threadIdx.x;
    const int m = lane & 15;
    const int h = lane >> 4;
    const int l0 = blockIdx.x * 64;
    const int t0 = blockIdx.y * 16;

    int tq = t0 + m; if (tq > TT - 1) tq = TT - 1;   // clamp (data only, no EXEC change)
    const float* qrow = lq + (size_t)tq * DD;
    const float* kvb = kv + (size_t)b * L_KV * DD;
    const float* k0 = kvb + (size_t)(l0 +  0 + m) * DD;
    const float* k1 = kvb + (size_t)(l0 + 16 + m) * DD;
    const float* k2 = kvb + (size_t)(l0 + 32 + m) * DD;
    const float* k3 = kvb + (size_t)(l0 + 48 + m) * DD;

    v8f acc0 = {}, acc1 = {}, acc2 = {}, acc3 = {};
    for (int kb = 0; kb < DD; kb += 4) {
        const int ko = kb + 2 * h;
        v2f a  = *(const v2f*)(qrow + ko);
        v2f q0 = *(const v2f*)(k0 + ko);
        v2f q1 = *(const v2f*)(k1 + ko);
        v2f q2 = *(const v2f*)(k2 + ko);
        v2f q3 = *(const v2f*)(k3 + ko);
        WMMA_F32(acc0, a, q0);
        WMMA_F32(acc1, a, q1);
        WMMA_F32(acc2, a, q2);
        WMMA_F32(acc3, a, q3);
    }

    const float scale = 1.0f / sqrtf(2048.0f);
#pragma unroll
    for (int r = 0; r < 8; ++r) {
        const int t = t0 + r + 8 * h;
        if (t >= TT) continue;
        const int g = t / 243;
        const float qy = (float)(t / 27) * (1.0f / 26.0f);
        const float qx = (float)(t % 27) * (1.0f / 26.0f);
        const float lo = (float)(9 * g) / 27.0f;
        const float hi = (float)(9 * (g + 1)) / 27.0f;
        float* srow = scores + (size_t)t * L_KV;
        const float av[4] = {acc0[r], acc1[r], acc2[r], acc3[r]};
#pragma unroll
        for (int s4 = 0; s4 < 4; ++s4) {
            const int l = l0 + s4 * 16 + m;
            srow[l] = av[s4] * scale + spatial_bias(t, l, qy, qx, lo, hi);
        }
    }
}

// ---------------------------------------------------------------------------
// Kernel 3 (per batch): in-place softmax over l (1600) for each of 729 rows.
// One wave per row; 8 waves per block. grid = 92, block = 256.
// ---------------------------------------------------------------------------
__global__ __launch_bounds__(256)
void softmax_kernel(float* __restrict__ scores)
{
    const int wave = threadIdx.x >> 5;
    const int lane = threadIdx.x & 31;
    const int row = blockIdx.x * 8 + wave;
    if (row >= TT) return;
    float* s = scores + (size_t)row * L_KV;

    float mx = -3.4e38f;
    for (int i = lane; i < L_KV; i += 32) mx = fmaxf(mx, s[i]);
#pragma unroll
    for (int off = 16; off > 0; off >>= 1) mx = fmaxf(mx, __shfl_xor(mx, off, 32));

    float sum = 0.0f;
    for (int i = lane; i < L_KV; i += 32) {
        const float e = expf(s[i] - mx);
        s[i] = e;
        sum += e;
    }
#pragma unroll
    for (int off = 16; off > 0; off >>= 1) sum += __shfl_xor(sum, off, 32);

    const float inv = 1.0f / sum;
    for (int i = lane; i < L_KV; i += 32) s[i] *= inv;
}

// ---------------------------------------------------------------------------
// Kernel 4 (per batch): out[t,d] = sum_l weights[t,l] * kv[b,l,d]
// A = weights (row-major, K=l contiguous); B = kv rows (column loads, stride DD).
// Each wave: 16(t) x 64(d). grid = (DD/64=32, 46), block = 32
// ---------------------------------------------------------------------------
__global__ __launch_bounds__(32)
void out_kernel(const float* __restrict__ weights, const float* __restrict__ kv,
                float* __restrict__ out, int b)
{
    const int lane = threadIdx.x;
    const int m = lane & 15;
    const int h = lane >> 4;
    const int n0 = blockIdx.x * 64;
    const int t0 = blockIdx.y * 16;

    int tq = t0 + m; if (tq > TT - 1) tq = TT - 1;
    const float* wrow = weights + (size_t)tq * L_KV;
    const float* kvb = kv + (size_t)b * L_KV * DD;

    v8f acc0 = {}, acc1 = {}, acc2 = {}, acc3 = {};
    for (int kb = 0; kb < L_KV; kb += 4) {
        const int ko = kb + 2 * h;
        v2f a = *(const v2f*)(wrow + ko);
        const float* kr0 = kvb + (size_t)ko * DD + n0 + m;   // kv row ko
        const float* kr1 = kr0 + DD;                         // kv row ko+1
        v2f q0, q1, q2, q3;
        q0[0] = kr0[ 0]; q0[1] = kr1[ 0];
        q1[0] = kr0[16]; q1[1] = kr1[16];
        q2[0] = kr0[32]; q2[1] = kr1[32];
        q3[0] = kr0[48]; q3[1] = kr1[48];
        WMMA_F32(acc0, a, q0);
        WMMA_F32(acc1, a, q1);
        WMMA_F32(acc2, a, q2);
        WMMA_F32(acc3, a, q3);
    }

    float* ob = out + (size_t)b * TT * DD;
#pragma unroll
    for (int r = 0; r < 8; ++r) {
        const int t = t0 + r + 8 * h;
        if (t >= TT) continue;
        float* orow = ob + (size_t)t * DD;
        orow[n0 +  0 + m] = acc0[r];
        orow[n0 + 16 + m] = acc1[r];
        orow[n0 + 32 + m] = acc2[r];
        orow[n0 + 48 + m] = acc3[r];
    }
}

// ---------------------------------------------------------------------------
extern "C" void kernel_launch(void* const* d_in, const int* in_sizes, int n_in,
                              void* d_out, int out_size, void* d_ws, size_t ws_size,
                              hipStream_t stream)
{
    (void)in_sizes; (void)n_in; (void)out_size; (void)ws_size;
    const float* x0 = (const float*)d_in[0];   // (8,576,1024)
    const float* x1 = (const float*)d_in[1];   // (8,1024,1536)
    const float* W0 = (const float*)d_in[2];   // (2048,1024)
    const float* b0 = (const float*)d_in[3];   // (2048,)
    const float* W1 = (const float*)d_in[4];   // (2048,1536)
    const float* b1 = (const float*)d_in[5];   // (2048,)
    const float* lq = (const float*)d_in[6];   // (1,729,2048)
    float* out = (float*)d_out;                // (8,729,2048)

    // Workspace: kv (8*1600*2048 f32 = 104.9 MB) + one batch of scores (729*1600 f32)
    float* kv = (float*)d_ws;
    float* scores = kv + (size_t)BB * L_KV * DD;

    kv_proj_kernel<<<dim3(DD / 64, L_KV / 16, BB), 32, 0, stream>>>(
        x0, x1, W0, b0, W1, b1, kv);

    for (int b = 0; b < BB; ++b) {
        scores_kernel<<<dim3(L_KV / 64, (TT + 15) / 16), 32, 0, stream>>>(lq, kv, scores, b);
        softmax_kernel<<<dim3((TT + 7) / 8), 256, 0, stream>>>(scores);
        out_kernel<<<dim3(DD / 64, (TT + 15) / 16), 32, 0, stream>>>(scores, kv, out, b);
    }
}